// RecModel_36429912604943
// MI455X (gfx1250) — compile-verified
//
#include <hip/hip_runtime.h>
#include <hip/hip_bf16.h>

// ---------------- problem constants ----------------
constexpr int N_ENT = 30000, N_CON = 29309, DIM = 128, NB = 8;
constexpr int BSZ = 256, LC = 50, SS = 32, E_DB = 300000, E_CON = 200000;

// ---------------- workspace layout (float offsets) ----------------
constexpr size_t OFF_AGG  = 0;                                   // N_ENT*DIM (-> db_feats)
constexpr size_t OFF_CNT  = OFF_AGG  + (size_t)N_ENT * DIM;      // N_ENT
constexpr size_t OFF_CONF = OFF_CNT  + N_ENT;                    // N_CON*DIM (con_feats accum)
constexpr size_t OFF_EDB  = OFF_CONF + (size_t)N_CON * DIM;      // BSZ*SS
constexpr size_t OFF_ECON = OFF_EDB  + (size_t)BSZ * SS;         // BSZ*LC
constexpr size_t OFF_SQ   = OFF_ECON + (size_t)BSZ * LC;         // BSZ
constexpr size_t OFF_SCAL = OFF_SQ   + BSZ;                      // 8 (scal[0] = rec_loss acc)
constexpr size_t ZERO_END = OFF_SCAL + 8;                        // everything above memset to 0
constexpr size_t OFF_DEG  = ZERO_END;                            // N_CON (filled 1.0 = self loop)
constexpr size_t OFF_XW   = OFF_DEG  + N_CON;                    // N_CON*DIM
constexpr size_t OFF_H    = OFF_XW   + (size_t)N_CON * DIM;      // BSZ*SS*DIM
constexpr size_t OFF_HC   = OFF_H    + (size_t)BSZ * SS * DIM;   // BSZ*LC*DIM
constexpr size_t OFF_DBU  = OFF_HC   + (size_t)BSZ * LC * DIM;   // BSZ*DIM
constexpr size_t OFF_CONU = OFF_DBU  + (size_t)BSZ * DIM;        // BSZ*DIM
constexpr size_t OFF_CAT  = OFF_CONU + (size_t)BSZ * DIM;        // BSZ*2*DIM
constexpr size_t OFF_U    = OFF_CAT  + (size_t)BSZ * 2 * DIM;    // BSZ*DIM
constexpr size_t OFF_UEMB = OFF_U    + (size_t)BSZ * DIM;        // BSZ*DIM
constexpr size_t OFF_CONE = OFF_UEMB + (size_t)BSZ * DIM;        // BSZ*DIM
constexpr size_t OFF_WT0  = OFF_CONE + (size_t)BSZ * DIM;        // DIM*DIM (gcn_w^T)
constexpr size_t OFF_WT1  = OFF_WT0  + (size_t)DIM * DIM;        // DIM*DIM (attn_a_db^T)
constexpr size_t OFF_WT2  = OFF_WT1  + (size_t)DIM * DIM;        // DIM*DIM (attn_a^T)

typedef __attribute__((ext_vector_type(2))) float v2f;
typedef __attribute__((ext_vector_type(8))) float v8f;

__device__ inline v8f wmma4(v2f a, v2f b, v8f c) {
  return __builtin_amdgcn_wmma_f32_16x16x4_f32(false, a, false, b, (short)0, c,
                                               false, false);
}

// ---------------- generic single-tile fp32 WMMA GEMM (BT layout only) ----
// B stored row-major [N x ldb]; element B[k][n] = Bm[n*ldb + k].
// Loads clamp m/n to M-1/N-1; stores are guarded by real row/col.
template <bool BIAS>
__global__ void k_gemm(const float* __restrict__ A, int lda,
                       const float* __restrict__ Bm, int ldb,
                       const float* __restrict__ bias, float* __restrict__ C,
                       int ldc, int M, int N, int K, int ntiles, int ntn) {
  int wid = blockIdx.x * (blockDim.x >> 5) + (threadIdx.x >> 5);
  if (wid >= ntiles) return;
  int tm = wid / ntn, tn = wid % ntn;
  int lane = threadIdx.x & 31;
  int r = lane & 15, half = lane >> 4;
  int mc = min(tm * 16 + r, M - 1);
  int nc = min(tn * 16 + r, N - 1);
  const float* ap = A + (size_t)mc * lda + half * 2;
  const float* bp = Bm + (size_t)nc * ldb + half * 2;
  v8f c = {};
#pragma unroll 8
  for (int k0 = 0; k0 < K; k0 += 4) {
    v2f a = *(const v2f*)ap;
    v2f b = *(const v2f*)bp;
    c = wmma4(a, b, c);
    ap += 4;
    bp += 4;
  }
  int col = tn * 16 + r;
  float bc = BIAS ? bias[nc] : 0.f;
#pragma unroll
  for (int j = 0; j < 8; ++j) {
    int row = tm * 16 + j + half * 8;
    if (row < M && col < N) C[(size_t)row * ldc + col] = c[j] + bc;
  }
}

// ---------------- NT-wide N-blocked GEMM (BT layout), bias epilogue ----
// One wave owns a 16x(16*NT) strip; A fragment reused NT times. Single base
// pointer + immediate per-tile offsets (t*16*K floats) so loads clause/pipeline.
// Requires M % 16 == 0 and N % (16*NT) == 0.
template <int NT>
__global__ void k_gemm_bt_multi(const float* __restrict__ A,
                                const float* __restrict__ Bm,
                                const float* __restrict__ bias,
                                float* __restrict__ C, int N, int K,
                                int nwaves, int ngn) {
  int wid = blockIdx.x * (blockDim.x >> 5) + (threadIdx.x >> 5);
  if (wid >= nwaves) return;
  int tm = wid / ngn, tg = wid % ngn;
  int lane = threadIdx.x & 31;
  int r = lane & 15, half = lane >> 4;
  const float* ap = A + (size_t)(tm * 16 + r) * K + half * 2;
  const float* bbase = Bm + (size_t)(tg * NT * 16 + r) * K + half * 2;
  v8f zero = {};
  v8f acc[NT];
#pragma unroll
  for (int t = 0; t < NT; ++t) acc[t] = zero;
#pragma unroll 4
  for (int k0 = 0; k0 < K; k0 += 4) {
    v2f a = *(const v2f*)ap;
    ap += 4;
    v2f b[NT];
#pragma unroll
    for (int t = 0; t < NT; ++t) b[t] = *(const v2f*)(bbase + (size_t)t * 16 * K);
    bbase += 4;
#pragma unroll
    for (int t = 0; t < NT; ++t) acc[t] = wmma4(a, b[t], acc[t]);
  }
#pragma unroll
  for (int t = 0; t < NT; ++t) {
    int colt = (tg * NT + t) * 16 + r;
    float bc = bias[colt];
#pragma unroll
    for (int j = 0; j < 8; ++j)
      C[(size_t)(tm * 16 + j + half * 8) * N + colt] = acc[t][j] + bc;
  }
}

// ---------------- NT-wide N-blocked GEMM + squared-error row reduction ----
// sq[row] += sum over this wave's 16*NT cols of ((A@B^T)+bias-dbv)^2
template <int NT>
__global__ void k_gemm_sq_multi(const float* __restrict__ A,
                                const float* __restrict__ Bm,
                                const float* __restrict__ bias,
                                const float* __restrict__ dbv,
                                float* __restrict__ sq, int N, int K,
                                int nwaves, int ngn) {
  int wid = blockIdx.x * (blockDim.x >> 5) + (threadIdx.x >> 5);
  if (wid >= nwaves) return;
  int tm = wid / ngn, tg = wid % ngn;
  int lane = threadIdx.x & 31;
  int r = lane & 15, half = lane >> 4;
  const float* ap = A + (size_t)(tm * 16 + r) * K + half * 2;
  const float* bbase = Bm + (size_t)(tg * NT * 16 + r) * K + half * 2;
  v8f zero = {};
  v8f acc[NT];
#pragma unroll
  for (int t = 0; t < NT; ++t) acc[t] = zero;
#pragma unroll 4
  for (int k0 = 0; k0 < K; k0 += 4) {
    v2f a = *(const v2f*)ap;
    ap += 4;
    v2f b[NT];
#pragma unroll
    for (int t = 0; t < NT; ++t) b[t] = *(const v2f*)(bbase + (size_t)t * 16 * K);
    bbase += 4;
#pragma unroll
    for (int t = 0; t < NT; ++t) acc[t] = wmma4(a, b[t], acc[t]);
  }
  float bc[NT];
  int colt[NT];
#pragma unroll
  for (int t = 0; t < NT; ++t) {
    colt[t] = (tg * NT + t) * 16 + r;
    bc[t] = bias[colt[t]];
  }
#pragma unroll
  for (int j = 0; j < 8; ++j) {
    int row = tm * 16 + j + half * 8;
    float v = 0.f;
#pragma unroll
    for (int t = 0; t < NT; ++t) {
      float d = acc[t][j] + bc[t] - dbv[(size_t)row * N + colt[t]];
      v += d * d;
    }
    v += __shfl_xor(v, 8, 32);
    v += __shfl_xor(v, 4, 32);
    v += __shfl_xor(v, 2, 32);
    v += __shfl_xor(v, 1, 32);
    if (r == 0) atomicAdd(&sq[row], v);
  }
}

// ---------------- full-width tanh projection: e[row] = sum_col tanh(X@Wt^T)*vec
// Wt is the transposed weight [N x K] so B loads are contiguous b64.
// One wave computes the whole 16x128 strip; reduction fully in-wave.
__global__ void k_tanh_full(const float* __restrict__ A, const float* __restrict__ Wt,
                            const float* __restrict__ vec, float* __restrict__ e,
                            int nwaves) {
  int tm = blockIdx.x * (blockDim.x >> 5) + (threadIdx.x >> 5);
  if (tm >= nwaves) return;
  int lane = threadIdx.x & 31;
  int r = lane & 15, half = lane >> 4;
  const float* ap = A + (size_t)(tm * 16 + r) * DIM + half * 2;
  const float* bbase = Wt + (size_t)r * DIM + half * 2;
  v8f zero = {};
  v8f acc[8];
#pragma unroll
  for (int t = 0; t < 8; ++t) acc[t] = zero;
#pragma unroll 4
  for (int k0 = 0; k0 < DIM; k0 += 4) {
    v2f a = *(const v2f*)ap;
    ap += 4;
    v2f b[8];
#pragma unroll
    for (int t = 0; t < 8; ++t) b[t] = *(const v2f*)(bbase + (size_t)t * 16 * DIM);
    bbase += 4;
#pragma unroll
    for (int t = 0; t < 8; ++t) acc[t] = wmma4(a, b[t], acc[t]);
  }
  float vv[8];
#pragma unroll
  for (int t = 0; t < 8; ++t) vv[t] = vec[t * 16 + r];
#pragma unroll
  for (int j = 0; j < 8; ++j) {
    float v = 0.f;
#pragma unroll
    for (int t = 0; t < 8; ++t) v += tanhf(acc[t][j]) * vv[t];
    v += __shfl_xor(v, 8, 32);
    v += __shfl_xor(v, 4, 32);
    v += __shfl_xor(v, 2, 32);
    v += __shfl_xor(v, 1, 32);
    if (r == 0) e[tm * 16 + j + half * 8] = v;
  }
}

// ---------------- small utility kernels ----------------
__global__ void k_fill(float* p, float v, int n) {
  int i = blockIdx.x * blockDim.x + threadIdx.x;
  if (i < n) p[i] = v;
}

// transpose a DIM x DIM matrix: outp[n*DIM + k] = in[k*DIM + n]
__global__ void k_transpose128(const float* __restrict__ in, float* __restrict__ outp) {
  int i = blockIdx.x * blockDim.x + threadIdx.x;
  if (i >= DIM * DIM) return;
  int k = i / DIM, n = i % DIM;
  outp[(size_t)n * DIM + k] = in[i];
}

// RGCN edge scatter: one wave per edge; msg = sum_b comp[rel,b]*basis[b,src,:]
__global__ void k_rgcn_edges(const int* __restrict__ ei, const int* __restrict__ et,
                             const float* __restrict__ basis,
                             const float* __restrict__ comp,
                             float* __restrict__ agg, float* __restrict__ cnt) {
  int e = blockIdx.x * (blockDim.x >> 5) + (threadIdx.x >> 5);
  if (e >= E_DB) return;
  int lane = threadIdx.x & 31;
  int src = ei[e], dst = ei[E_DB + e], rel = et[e];
  int d4 = lane * 4;
  float4 m = {0.f, 0.f, 0.f, 0.f};
#pragma unroll
  for (int b = 0; b < NB; ++b) {
    float cb = comp[rel * NB + b];
    float4 v = *reinterpret_cast<const float4*>(basis + ((size_t)b * N_ENT + src) * DIM + d4);
    m.x += cb * v.x; m.y += cb * v.y; m.z += cb * v.z; m.w += cb * v.w;
  }
  float* p = agg + (size_t)dst * DIM + d4;
  atomicAdd(p + 0, m.x); atomicAdd(p + 1, m.y);
  atomicAdd(p + 2, m.z); atomicAdd(p + 3, m.w);
  if (lane == 0) atomicAdd(&cnt[dst], 1.0f);
}

__global__ void k_rgcn_fin(float* __restrict__ feats, const float* __restrict__ cnt,
                           const float* __restrict__ root, const float* __restrict__ bias) {
  size_t i = (size_t)blockIdx.x * blockDim.x + threadIdx.x;
  if (i >= (size_t)N_ENT * DIM) return;
  int n = (int)(i / DIM), d = (int)(i % DIM);
  feats[i] = feats[i] / fmaxf(cnt[n], 1.0f) + root[i] + bias[d];
}

__global__ void k_con_deg(const int* __restrict__ cei, float* __restrict__ deg) {
  int e = blockIdx.x * blockDim.x + threadIdx.x;
  if (e < E_CON) atomicAdd(&deg[cei[E_CON + e]], 1.0f);
}

__global__ void k_gcn_edges(const int* __restrict__ cei, const float* __restrict__ deg,
                            const float* __restrict__ xw, float* __restrict__ conf) {
  int e = blockIdx.x * (blockDim.x >> 5) + (threadIdx.x >> 5);
  if (e >= E_CON) return;
  int lane = threadIdx.x & 31;
  int s = cei[e], d = cei[E_CON + e];
  float norm = rsqrtf(deg[s]) * rsqrtf(deg[d]);
  int d4 = lane * 4;
  float4 v = *reinterpret_cast<const float4*>(xw + (size_t)s * DIM + d4);
  float* p = conf + (size_t)d * DIM + d4;
  atomicAdd(p + 0, norm * v.x); atomicAdd(p + 1, norm * v.y);
  atomicAdd(p + 2, norm * v.z); atomicAdd(p + 3, norm * v.w);
}

__global__ void k_gcn_fin(float* __restrict__ conf, const float* __restrict__ xw,
                          const float* __restrict__ deg, const float* __restrict__ gcn_b) {
  size_t i = (size_t)blockIdx.x * blockDim.x + threadIdx.x;
  if (i >= (size_t)N_CON * DIM) return;
  int n = (int)(i / DIM), d = (int)(i % DIM);
  conf[i] = conf[i] + xw[i] / deg[n] + gcn_b[d];  // self-loop: dinv[n]^2 = 1/deg[n]
}

__global__ void k_gather(const float* __restrict__ src, const int* __restrict__ idx,
                         float* __restrict__ dst, int rows) {
  size_t i = (size_t)blockIdx.x * blockDim.x + threadIdx.x;
  if (i >= (size_t)rows * DIM) return;
  int r = (int)(i / DIM), d = (int)(i % DIM);
  dst[i] = src[(size_t)idx[r] * DIM + d];
}

// softmax over S=32 seeds (mask pos<len) then weighted sum of h -> db_user
__global__ void k_attn_db(const float* __restrict__ e, const float* __restrict__ h,
                          const int* __restrict__ lens, float* __restrict__ dbu) {
  __shared__ float w[SS];
  int b = blockIdx.x, t = threadIdx.x;  // 128 threads
  int len = lens[b];
  if (t == 0) {
    float mx = -3.4e38f;
    for (int s = 0; s < SS; ++s)
      if (s < len) mx = fmaxf(mx, e[b * SS + s]);
    float sum = 0.f;
    for (int s = 0; s < SS; ++s) {
      float ws = (s < len) ? expf(e[b * SS + s] - mx) : 0.f;
      w[s] = ws; sum += ws;
    }
    float inv = (sum > 0.f) ? 1.f / sum : 0.f;
    for (int s = 0; s < SS; ++s) w[s] *= inv;
  }
  __syncthreads();
  float acc = 0.f;
  for (int s = 0; s < SS; ++s) acc += w[s] * h[((size_t)b * SS + s) * DIM + t];
  dbu[b * DIM + t] = (len > 0) ? acc : 0.f;
}

// softmax over LC=50 concepts (mask id!=0; all-invalid -> uniform) -> con_user
__global__ void k_attn_con(const float* __restrict__ e, const float* __restrict__ hc,
                           const int* __restrict__ cm, float* __restrict__ conu) {
  __shared__ float w[LC];
  int b = blockIdx.x, t = threadIdx.x;  // 128 threads
  if (t == 0) {
    int nv = 0; float mx = -3.4e38f;
    for (int s = 0; s < LC; ++s)
      if (cm[b * LC + s] != 0) { ++nv; mx = fmaxf(mx, e[b * LC + s]); }
    if (nv == 0) {
      for (int s = 0; s < LC; ++s) w[s] = 1.f / LC;
    } else {
      float sum = 0.f;
      for (int s = 0; s < LC; ++s) {
        float ws = (cm[b * LC + s] != 0) ? expf(e[b * LC + s] - mx) : 0.f;
        w[s] = ws; sum += ws;
      }
      float inv = 1.f / sum;
      for (int s = 0; s < LC; ++s) w[s] *= inv;
    }
  }
  __syncthreads();
  float acc = 0.f;
  for (int s = 0; s < LC; ++s) acc += w[s] * hc[((size_t)b * LC + s) * DIM + t];
  conu[b * DIM + t] = acc;
}

__global__ void k_cat(const float* __restrict__ conu, const float* __restrict__ dbu,
                      float* __restrict__ cat) {
  int i = blockIdx.x * blockDim.x + threadIdx.x;
  if (i >= BSZ * 2 * DIM) return;
  int b = i / (2 * DIM), d = i % (2 * DIM);
  cat[i] = (d < DIM) ? conu[b * DIM + d] : dbu[b * DIM + d - DIM];
}

// gate = sigmoid(u . gw + gb); user_emb = gate*db_user + (1-gate)*con_user
__global__ void k_gate(const float* __restrict__ u, const float* __restrict__ gw,
                       const float* __restrict__ gb, const float* __restrict__ dbu,
                       const float* __restrict__ conu, float* __restrict__ uemb) {
  __shared__ float sh[DIM];
  int b = blockIdx.x, t = threadIdx.x;  // 128 threads
  sh[t] = u[b * DIM + t] * gw[t];
  __syncthreads();
  for (int s = 64; s > 0; s >>= 1) {
    if (t < s) sh[t] += sh[t + s];
    __syncthreads();
  }
  float g = 1.f / (1.f + expf(-(sh[0] + gb[0])));
  uemb[b * DIM + t] = g * dbu[b * DIM + t] + (1.f - g) * conu[b * DIM + t];
}

// per-row log-softmax CE over entity_scores; rec_acc += ce*rec
__global__ void k_ce(const float* __restrict__ scores, const int* __restrict__ labels,
                     const float* __restrict__ rec, float* __restrict__ rec_acc) {
  __shared__ float sh[256];
  int b = blockIdx.x, t = threadIdx.x;
  const float* row = scores + (size_t)b * N_ENT;
  float mx = -3.4e38f;
  for (int i = t; i < N_ENT; i += 256) mx = fmaxf(mx, row[i]);
  sh[t] = mx; __syncthreads();
  for (int s = 128; s > 0; s >>= 1) {
    if (t < s) sh[t] = fmaxf(sh[t], sh[t + s]);
    __syncthreads();
  }
  mx = sh[0]; __syncthreads();
  float sum = 0.f;
  for (int i = t; i < N_ENT; i += 256) sum += expf(row[i] - mx);
  sh[t] = sum; __syncthreads();
  for (int s = 128; s > 0; s >>= 1) {
    if (t < s) sh[t] += sh[t + s];
    __syncthreads();
  }
  if (t == 0) {
    float ce = (mx + logf(sh[0])) - row[labels[b]];
    atomicAdd(rec_acc, ce * rec[b]);
  }
}

__global__ void k_final(const float* __restrict__ sq, const int* __restrict__ lens,
                        const float* __restrict__ scal, float* __restrict__ out_loss,
                        float* __restrict__ out_rec) {
  __shared__ float sh[BSZ];
  int t = threadIdx.x;
  sh[t] = sq[t] * (lens[t] > 0 ? 1.f : 0.f);
  __syncthreads();
  for (int s = 128; s > 0; s >>= 1) {
    if (t < s) sh[t] += sh[t + s];
    __syncthreads();
  }
  if (t == 0) {
    float rl = scal[0];
    *out_rec = rl;
    *out_loss = rl + 0.025f * (sh[0] / (float)BSZ);
  }
}

// ---------------- launcher ----------------
extern "C" void kernel_launch(void* const* d_in, const int* in_sizes, int n_in,
                              void* d_out, int out_size, void* d_ws, size_t ws_size,
                              hipStream_t stream) {
  (void)in_sizes; (void)n_in; (void)out_size; (void)ws_size;
  const int*   concept_mask = (const int*)d_in[0];
  const int*   seed_pad     = (const int*)d_in[1];
  const int*   seed_lens    = (const int*)d_in[2];
  const int*   db_ei        = (const int*)d_in[3];
  const int*   db_et        = (const int*)d_in[4];
  const int*   con_ei       = (const int*)d_in[5];
  const float* db_vec       = (const float*)d_in[6];
  const int*   labels       = (const int*)d_in[7];
  const float* rec          = (const float*)d_in[8];
  const float* concept_emb  = (const float*)d_in[9];
  const float* basis        = (const float*)d_in[10];
  const float* comp         = (const float*)d_in[11];
  const float* rgcn_root    = (const float*)d_in[12];
  const float* rgcn_bias    = (const float*)d_in[13];
  const float* gcn_w        = (const float*)d_in[14];
  const float* gcn_b        = (const float*)d_in[15];
  const float* attn_a       = (const float*)d_in[16];
  const float* attn_b       = (const float*)d_in[17];
  const float* attn_a_db    = (const float*)d_in[18];
  const float* attn_b_db    = (const float*)d_in[19];
  const float* unw          = (const float*)d_in[20];
  const float* unb          = (const float*)d_in[21];
  const float* gnw          = (const float*)d_in[22];
  const float* gnb          = (const float*)d_in[23];
  const float* icw          = (const float*)d_in[24];
  const float* icb          = (const float*)d_in[25];
  const float* iob          = (const float*)d_in[26];
  const float* oeb          = (const float*)d_in[27];

  float* out = (float*)d_out;
  float* ws  = (float*)d_ws;
  float* agg   = ws + OFF_AGG;    // becomes db_feats
  float* cnt   = ws + OFF_CNT;
  float* conf  = ws + OFF_CONF;   // becomes con_feats
  float* e_db  = ws + OFF_EDB;
  float* e_con = ws + OFF_ECON;
  float* sq    = ws + OFF_SQ;
  float* scal  = ws + OFF_SCAL;
  float* deg   = ws + OFF_DEG;
  float* xw    = ws + OFF_XW;
  float* h     = ws + OFF_H;
  float* hc    = ws + OFF_HC;
  float* dbu   = ws + OFF_DBU;
  float* conu  = ws + OFF_CONU;
  float* cat   = ws + OFF_CAT;
  float* u     = ws + OFF_U;
  float* uemb  = ws + OFF_UEMB;
  float* cone  = ws + OFF_CONE;
  float* gcn_wT    = ws + OFF_WT0;
  float* attn_adbT = ws + OFF_WT1;
  float* attn_aT   = ws + OFF_WT2;
  float* entity_scores = out + 1;                    // [BSZ, N_ENT]
  float* out_rec_loss  = out + 1 + (size_t)BSZ * N_ENT;

  // 0) zero accumulators, fill deg (self loop), transpose small weights
  hipMemsetAsync(ws, 0, ZERO_END * sizeof(float), stream);
  k_fill<<<(N_CON + 255) / 256, 256, 0, stream>>>(deg, 1.0f, N_CON);
  k_transpose128<<<(DIM * DIM + 255) / 256, 256, 0, stream>>>(gcn_w, gcn_wT);
  k_transpose128<<<(DIM * DIM + 255) / 256, 256, 0, stream>>>(attn_a_db, attn_adbT);
  k_transpose128<<<(DIM * DIM + 255) / 256, 256, 0, stream>>>(attn_a, attn_aT);

  // 1) RGCN: per-edge basis-decomposed messages -> mean aggregate
  k_rgcn_edges<<<(E_DB + 7) / 8, 256, 0, stream>>>(db_ei, db_et, basis, comp, agg, cnt);
  {
    size_t n = (size_t)N_ENT * DIM;
    k_rgcn_fin<<<(unsigned)((n + 255) / 256), 256, 0, stream>>>(agg, cnt, rgcn_root, rgcn_bias);
  }

  // 2) GCN: degrees, xw = concept_emb @ gcn_w (WMMA, BT vs gcn_w^T), scatter, finalize
  k_con_deg<<<(E_CON + 255) / 256, 256, 0, stream>>>(con_ei, deg);
  {
    int ntn = DIM / 16, ntm = (N_CON + 15) / 16, ntiles = ntm * ntn;
    k_gemm<false><<<(ntiles + 3) / 4, 128, 0, stream>>>(
        concept_emb, DIM, gcn_wT, DIM, nullptr, xw, DIM, N_CON, DIM, DIM, ntiles, ntn);
  }
  k_gcn_edges<<<(E_CON + 7) / 8, 256, 0, stream>>>(con_ei, deg, xw, conf);
  {
    size_t n = (size_t)N_CON * DIM;
    k_gcn_fin<<<(unsigned)((n + 255) / 256), 256, 0, stream>>>(conf, xw, deg, gcn_b);
  }

  // 3) gathers for attention
  k_gather<<<(BSZ * SS * DIM + 255) / 256, 256, 0, stream>>>(agg, seed_pad, h, BSZ * SS);
  k_gather<<<(BSZ * LC * DIM + 255) / 256, 256, 0, stream>>>(conf, concept_mask, hc, BSZ * LC);

  // 4) attention scores e = tanh(X@A)@b  (WMMA, full 16x128 strip per wave)
  {
    int nwaves = BSZ * SS / 16;  // 512
    k_tanh_full<<<(nwaves + 3) / 4, 128, 0, stream>>>(h, attn_adbT, attn_b_db, e_db, nwaves);
  }
  {
    int nwaves = BSZ * LC / 16;  // 800
    k_tanh_full<<<(nwaves + 3) / 4, 128, 0, stream>>>(hc, attn_aT, attn_b, e_con, nwaves);
  }
  k_attn_db<<<BSZ, DIM, 0, stream>>>(e_db, h, seed_lens, dbu);
  k_attn_con<<<BSZ, DIM, 0, stream>>>(e_con, hc, concept_mask, conu);

  // 5) gated fusion: u = cat @ unw^T + unb (WMMA), gate, user_emb
  k_cat<<<(BSZ * 2 * DIM + 255) / 256, 256, 0, stream>>>(conu, dbu, cat);
  {
    int ntn = DIM / 16, ntiles = (BSZ / 16) * ntn;
    k_gemm<true><<<(ntiles + 3) / 4, 128, 0, stream>>>(
        cat, 2 * DIM, unw, 2 * DIM, unb, u, DIM, BSZ, DIM, 2 * DIM, ntiles, ntn);
  }
  k_gate<<<BSZ, DIM, 0, stream>>>(u, gnw, gnb, dbu, conu, uemb);

  // 6) con_e = con_user @ icw^T + icb (WMMA)
  {
    int ntn = DIM / 16, ntiles = (BSZ / 16) * ntn;
    k_gemm<true><<<(ntiles + 3) / 4, 128, 0, stream>>>(
        conu, DIM, icw, DIM, icb, cone, DIM, BSZ, DIM, DIM, ntiles, ntn);
  }

  // 7) entity_scores = user_emb @ db_feats^T + oeb (WMMA, 5 N-tiles per wave)
  {
    int ngn = N_ENT / (16 * 5);                // 375 groups
    int nwaves = (BSZ / 16) * ngn;             // 6000
    k_gemm_bt_multi<5><<<(nwaves + 3) / 4, 128, 0, stream>>>(
        uemb, agg, oeb, entity_scores, N_ENT, DIM, nwaves, ngn);
  }

  // 8) infomax: sq[b] += rowsum((con_e @ db_feats^T + iob - db_vec)^2) (fused)
  {
    int ngn = N_ENT / (16 * 5);
    int nwaves = (BSZ / 16) * ngn;
    k_gemm_sq_multi<5><<<(nwaves + 3) / 4, 128, 0, stream>>>(
        cone, agg, iob, db_vec, sq, N_ENT, DIM, nwaves, ngn);
  }

  // 9) CE + final loss assembly
  k_ce<<<BSZ, 256, 0, stream>>>(entity_scores, labels, rec, &scal[0]);
  k_final<<<1, BSZ, 0, stream>>>(sq, seed_lens, scal, out, out_rec_loss);
}